// Kinematics_Physics_30064771072750
// MI455X (gfx1250) — compile-verified
//
#include <hip/hip_runtime.h>
#include <cstdint>
#include <cstddef>

typedef uint32_t v4u32 __attribute__((ext_vector_type(4)));
typedef int      v4i32 __attribute__((ext_vector_type(4)));
typedef int      v8i32 __attribute__((ext_vector_type(8)));

#define DEG2RADF 0.017453292519943295f
#define RAD2DEGF 57.29577951308232f
#define TINYF    1e-6f
#define BLOCK    256
#define TILE_ELEMS (BLOCK * 6u)   // 1536 floats = 6 KB per tile

// ---------------------------------------------------------------------------
// Kernel 0: fold the (6x4) DH table into per-joint constants in workspace:
//   ws[i*8 + {0..5}] = { theta_offset_deg, cos(alpha), sin(alpha), a,
//                        -sin(alpha)*d, cos(alpha)*d }
// ---------------------------------------------------------------------------
__global__ void fk_precompute(const float* __restrict__ dh, float* __restrict__ ws) {
    int i = threadIdx.x;
    if (i < 6) {
        float a   = dh[i * 4 + 0];
        float al  = dh[i * 4 + 1] * DEG2RADF;
        float d   = dh[i * 4 + 2];
        float off = dh[i * 4 + 3];
        float sa, ca;
        sincosf(al, &sa, &ca);
        float* w = ws + i * 8;
        w[0] = off;  w[1] = ca;       w[2] = sa;
        w[3] = a;    w[4] = -sa * d;  w[5] = ca * d;
        w[6] = 0.f;  w[7] = 0.f;
    }
}

// ---------------------------------------------------------------------------
// TDM tile DMA: 1-D tile of `TILE_ELEMS` f32 between LDS and global memory.
// `dim0` is the valid tensor extent (<= TILE_ELEMS): OOB reads return zero,
// OOB writes are dropped -> free tail handling.
// ---------------------------------------------------------------------------
template <bool STORE>
__device__ __forceinline__ void tdm_tile(uint32_t lds_off, uint64_t gaddr, uint32_t dim0) {
    // D# group 0: count=1, lds_addr, global_addr[56:0], type=2 ("image")
    v4u32 g0;
    g0.x = 1u;
    g0.y = lds_off;
    g0.z = (uint32_t)gaddr;
    g0.w = ((uint32_t)(gaddr >> 32) & 0x01FFFFFFu) | 0x80000000u;

    // D# group 1: data_size=4B (code 2); tensor_dim0=dim0; tensor_dim1=1;
    // tile_dim0=TILE_ELEMS; tile_dim1=1; tensor_dim0_stride=TILE_ELEMS
    v8i32 g1;
    g1[0] = (int)(2u << 16);                                 // data_size code 2 -> 4 bytes
    g1[1] = (int)((dim0 & 0xFFFFu) << 16);                   // tensor_dim0[15:0]
    g1[2] = (int)(((dim0 >> 16) & 0xFFFFu) | (1u << 16));    // tensor_dim0[31:16]; tensor_dim1=1
    g1[3] = (int)(TILE_ELEMS << 16);                         // tile_dim0
    g1[4] = 1;                                               // tile_dim1=1, tile_dim2=0
    g1[5] = (int)TILE_ELEMS;                                 // tensor_dim0_stride[31:0]
    g1[6] = 0;
    g1[7] = 0;

    v4i32 gz = {0, 0, 0, 0};
#if __clang_major__ >= 23
    v8i32 gz8 = {0, 0, 0, 0, 0, 0, 0, 0};
    if (STORE) __builtin_amdgcn_tensor_store_from_lds(g0, g1, gz, gz, gz8, 0);
    else       __builtin_amdgcn_tensor_load_to_lds (g0, g1, gz, gz, gz8, 0);
#else
    if (STORE) __builtin_amdgcn_tensor_store_from_lds(g0, g1, gz, gz, 0);
    else       __builtin_amdgcn_tensor_load_to_lds (g0, g1, gz, gz, 0);
#endif
}

// ---------------------------------------------------------------------------
// Kernel 1: one thread per batch element. Theta tile DMA'd in by the TDM,
// results staged in LDS and DMA'd out by the TDM (TENSORcnt-tracked;
// S_ENDPGM's implicit wait-idle drains the outstanding store).
// ---------------------------------------------------------------------------
__global__ __launch_bounds__(BLOCK)
void fk_main(const float* __restrict__ theta, const float* __restrict__ cst,
             float* __restrict__ out, int batch) {
    __shared__ float sTheta[TILE_ELEMS];
    __shared__ float sOut[TILE_ELEMS];
    const int      tid   = threadIdx.x;
    const uint32_t ebase = (uint32_t)blockIdx.x * TILE_ELEMS;  // float-element base
    const uint32_t total = (uint32_t)batch * 6u;
    uint32_t td0 = total - ebase;
    if (td0 > TILE_ELEMS) td0 = TILE_ELEMS;

    // ---- TDM in: one wave issues the async tensor load of this block's tile ----
    if (tid < 32) {
        tdm_tile<false>((uint32_t)(uintptr_t)&sTheta[0],
                        (uint64_t)(uintptr_t)theta + (uint64_t)ebase * 4ull, td0);
        __builtin_amdgcn_s_wait_tensorcnt(0);
    }
    __syncthreads();

    // ---- per-thread thetas from LDS (bank-conflict-free: 6*t mod 64 injective) ----
    float th[6];
#pragma unroll
    for (int i = 0; i < 6; ++i) th[i] = sTheta[tid * 6 + i];

    const float4* c4 = (const float4*)cst;

    // Joint 0 initializes T (3x4; last row implicit [0,0,0,1]).
    float T00, T01, T02, T03, T10, T11, T12, T13, T20, T21, T22, T23;
    {
        float4 kA = c4[0];   // off, ca, sa, a
        float4 kB = c4[1];   // -sa*d, ca*d, 0, 0
        float s, c;
        sincosf((kA.x + th[0]) * DEG2RADF, &s, &c);
        T00 = c;        T01 = -s;       T02 = 0.f;   T03 = kA.w;
        T10 = s * kA.y; T11 = c * kA.y; T12 = -kA.z; T13 = kB.x;
        T20 = s * kA.z; T21 = c * kA.z; T22 = kA.y;  T23 = kB.y;
    }

    // Joints 1..5: T = T * Ti, exploiting Ti's structure.
#pragma unroll
    for (int i = 1; i < 6; ++i) {
        float4 kA = c4[2 * i];
        float4 kB = c4[2 * i + 1];
        float s, c;
        sincosf((kA.x + th[i]) * DEG2RADF, &s, &c);
        float ca = kA.y, sa = kA.z, a = kA.w, msd = kB.x, cdd = kB.y;
        float sca = s * ca, cca = c * ca, ssa = s * sa, csa = c * sa;
        float n0, n1, n2, n3;

        n0 =  T00 * c + T01 * sca + T02 * ssa;
        n1 = -T00 * s + T01 * cca + T02 * csa;
        n2 =           -T01 * sa  + T02 * ca;
        n3 =  T00 * a + T01 * msd + T02 * cdd + T03;
        T00 = n0; T01 = n1; T02 = n2; T03 = n3;

        n0 =  T10 * c + T11 * sca + T12 * ssa;
        n1 = -T10 * s + T11 * cca + T12 * csa;
        n2 =           -T11 * sa  + T12 * ca;
        n3 =  T10 * a + T11 * msd + T12 * cdd + T13;
        T10 = n0; T11 = n1; T12 = n2; T13 = n3;

        n0 =  T20 * c + T21 * sca + T22 * ssa;
        n1 = -T20 * s + T21 * cca + T22 * csa;
        n2 =           -T21 * sa  + T22 * ca;
        n3 =  T20 * a + T21 * msd + T22 * cdd + T23;
        T20 = n0; T21 = n1; T22 = n2; T23 = n3;
    }

    // ---- Euler extraction (cosA*T00 - sinA*T01 == hypot(T00,T01)) ----
    float A1 = atan2f( T10, T11) * RAD2DEGF;
    float B1 = atan2f( T02, T22) * RAD2DEGF;
    float A2 = atan2f(-T01, T00) * RAD2DEGF;
    float B2 = atan2f( T02, sqrtf(T00 * T00 + T01 * T01)) * RAD2DEGF;
    float C2 = atan2f(-T12, T22) * RAD2DEGF;
    bool cond = (fabsf(T12) <= TINYF) && (fabsf(T22) <= TINYF);
    float A = cond ? A1 : A2;
    float B = cond ? B1 : B2;
    float C = cond ? 0.f : C2;

    // ---- stage results in LDS, then TDM out (OOB lanes dropped by descriptor) ----
    const int o = tid * 6;
    sOut[o + 0] = T03;  sOut[o + 1] = T13;  sOut[o + 2] = T23;
    sOut[o + 3] = A;    sOut[o + 4] = B;    sOut[o + 5] = C;
    __syncthreads();

    if (tid < 32) {
        tdm_tile<true>((uint32_t)(uintptr_t)&sOut[0],
                       (uint64_t)(uintptr_t)out + (uint64_t)ebase * 4ull, td0);
        // No explicit s_wait_tensorcnt needed: S_ENDPGM performs an implicit
        // wait-idle, and LDS persists until every wave (incl. this one) retires.
    }
}

// ---------------------------------------------------------------------------
extern "C" void kernel_launch(void* const* d_in, const int* in_sizes, int n_in,
                              void* d_out, int out_size, void* d_ws, size_t ws_size,
                              hipStream_t stream) {
    const float* theta = (const float*)d_in[0];   // (BATCH, 6) f32
    const float* dh    = (const float*)d_in[1];   // (6, 4)    f32
    float*       out   = (float*)d_out;           // (BATCH, 6) f32
    float*       ws    = (float*)d_ws;            // >= 48 floats used

    int batch = in_sizes[0] / 6;

    fk_precompute<<<1, 32, 0, stream>>>(dh, ws);

    int nblk = (batch + BLOCK - 1) / BLOCK;
    fk_main<<<nblk, BLOCK, 0, stream>>>(theta, ws, out, batch);
}